// SocialGroupLoss_23553600651773
// MI455X (gfx1250) — compile-verified
//
#include <hip/hip_runtime.h>

typedef __attribute__((ext_vector_type(2))) float v2f;
typedef __attribute__((ext_vector_type(8))) float v8f;

namespace {
constexpr int   kB = 64;    // batch (pairwise-distance dimension)
constexpr int   kN = 64;    // agents
constexpr int   kT = 100;   // timesteps
constexpr float kSocialDist = 1.5f;
constexpr int   kWaves = 2; // waves per block, each privately handles t = wave, wave+2, ...
// per-wave LDS slice (floats): Xs[64] | Ys[64] | Sq[64] | D2[64*64]
constexpr int   kSliceF = 64 * 3 + 64 * 64;
}

__launch_bounds__(kWaves * 32)
__global__ void social_group_loss_kernel(const float* __restrict__ x,
                                         const float* __restrict__ wfa,
                                         const int*   __restrict__ rand_idx,
                                         const unsigned char* __restrict__ drop_mask,
                                         float* __restrict__ out) {
  __shared__ float lds[kWaves * kSliceF];
  __shared__ float accbuf[kB];

  const int  tid  = threadIdx.x;
  const int  wave = tid >> 5;
  const int  lane = tid & 31;
  const int  l15  = lane & 15;
  // lanes 0-15 carry K=0,1 data; lanes 16-31 are the K=2,3 zero pad.
  // Use a multiplicative mask (not a ternary) so LDS loads stay unpredicated.
  const float keep = (lane < 16) ? 1.0f : 0.0f;
  const int  mhi   = (lane < 16) ? 0 : 8;   // D tile: vgpr r -> M=r (lo lanes) / r+8 (hi lanes)
  const int  n     = blockIdx.x;

  float* Xs = lds + wave * kSliceF;
  float* Ys = Xs + kB;
  float* Sq = Ys + kB;
  float* D2 = Sq + kB;                      // clamped squared distances, diag = +inf

  const int b0 = lane, b1 = lane + 32;      // each lane owns two batch rows
  const float kInf = __builtin_inff();

  // world_from_agent rows (loop invariant): pos_g[i] = R[i,0]*px + R[i,1]*py + t[i]
  const float R00a = wfa[b0 * 9 + 0], R01a = wfa[b0 * 9 + 1], Txa = wfa[b0 * 9 + 2];
  const float R10a = wfa[b0 * 9 + 3], R11a = wfa[b0 * 9 + 4], Tya = wfa[b0 * 9 + 5];
  const float R00b = wfa[b1 * 9 + 0], R01b = wfa[b1 * 9 + 1], Txb = wfa[b1 * 9 + 2];
  const float R10b = wfa[b1 * 9 + 3], R11b = wfa[b1 * 9 + 4], Tyb = wfa[b1 * 9 + 5];

  float acc0 = 0.0f, acc1 = 0.0f;

  for (int t = wave; t < kT; t += kWaves) {
    // ---- load + transform the 64 agent positions for this (n, t) ----
    const size_t off0 = ((size_t)(b0 * kN + n) * kT + t) * 6;
    const size_t off1 = ((size_t)(b1 * kN + n) * kT + t) * 6;
    const float2 p0 = *(const float2*)(x + off0);   // 8B aligned: offset is even*4B
    const float2 p1 = *(const float2*)(x + off1);
    if (t + kWaves < kT) {  // pull next iteration's strided line into cache
      __builtin_prefetch(x + off0 + 6 * kWaves, 0, 0);
      __builtin_prefetch(x + off1 + 6 * kWaves, 0, 0);
    }
    const float X0 = R00a * p0.x + R01a * p0.y + Txa;
    const float Y0 = R10a * p0.x + R11a * p0.y + Tya;
    const float X1 = R00b * p1.x + R01b * p1.y + Txb;
    const float Y1 = R10b * p1.x + R11b * p1.y + Tyb;

    Xs[b0] = X0; Ys[b0] = Y0; Sq[b0] = X0 * X0 + Y0 * Y0;
    Xs[b1] = X1; Ys[b1] = Y1; Sq[b1] = X1 * X1 + Y1 * Y1;
    // Same-wave LDS ops are in-order (DScnt); slice is wave-private -> no barrier.

    // ---- 64x64 Gram via V_WMMA_F32_16X16X4_F32 (K=2 padded to 4) ----
    // Store clamped d2 (diag=inf); sqrt is monotonic, so argmin can run in d2-space
    // and we only take sqrt for the finally-selected neighbor (2 per lane per t).
#pragma unroll
    for (int mi = 0; mi < 4; ++mi) {
      const int m = mi * 16 + l15;
      v2f a;                         // A 16x4: v0 = K0|K2, v1 = K1|K3 (lane halves)
      a.x = Xs[m] * keep;
      a.y = Ys[m] * keep;
      const int Mbase = mi * 16 + mhi;
#pragma unroll
      for (int ni = 0; ni < 4; ++ni) {
        const int nn = ni * 16 + l15;
        v2f bm;                      // B 4x16, mirrored K packing
        bm.x = Xs[nn] * keep;
        bm.y = Ys[nn] * keep;
        v8f c = {};
        v8f d = __builtin_amdgcn_wmma_f32_16x16x4_f32(
            /*neg_a=*/false, a, /*neg_b=*/false, bm,
            /*c_mod=*/(short)0, c, /*reuse_a=*/false, /*reuse_b=*/false);
        const int Ng  = ni * 16 + l15;
        const float sqn = Sq[Ng];
#pragma unroll
        for (int r = 0; r < 8; ++r) {
          const int Mg = Mbase + r;
          float d2 = Sq[Mg] + sqn - 2.0f * d[r];
          d2 = fmaxf(d2, 1e-12f);
          d2 = (Mg == Ng) ? kInf : d2;          // branchless diagonal mask
          D2[Mg * kB + Ng] = d2;
        }
      }
    }

    // ---- per-row argmin (in d2-space), neighbor select, loss accumulation ----
    const int rowbase = (t * kN + n) * kB;

    float best0 = kInf, best1 = kInf;
    int   bi0 = 0, bi1 = 0;
    for (int cidx = 0; cidx < kB; ++cidx) {     // strict '<' == first-occurrence argmin
      const float v0 = D2[b0 * kB + cidx];
      const float v1 = D2[b1 * kB + cidx];
      if (v0 < best0) { best0 = v0; bi0 = cidx; }
      if (v1 < best1) { best1 = v1; bi1 = cidx; }
    }
    const int r0 = rand_idx[rowbase + b0];
    const int r1 = rand_idx[rowbase + b1];
    const int rnb0 = r0 + (r0 >= b0 ? 1 : 0);   // skip-self random neighbor
    const int rnb1 = r1 + (r1 >= b1 ? 1 : 0);
    const int nb0 = drop_mask[rowbase + b0] ? rnb0 : bi0;
    const int nb1 = drop_mask[rowbase + b1] ? rnb1 : bi1;
    const float nd0 = __builtin_amdgcn_sqrtf(D2[b0 * kB + nb0]);  // raw v_sqrt_f32
    const float nd1 = __builtin_amdgcn_sqrtf(D2[b1 * kB + nb1]);
    const float e0 = nd0 - kSocialDist;
    const float e1 = nd1 - kSocialDist;
    acc0 += e0 * e0;
    acc1 += e1 * e1;
  }

  // combine the two waves' partials (fixed order -> deterministic), write loss[b][n]
  if (wave == 1) { accbuf[b0] = acc0; accbuf[b1] = acc1; }
  __syncthreads();
  if (wave == 0) {
    out[b0 * kN + n] = (acc0 + accbuf[b0]) * (1.0f / kT);
    out[b1 * kN + n] = (acc1 + accbuf[b1]) * (1.0f / kT);
  }
}

extern "C" void kernel_launch(void* const* d_in, const int* in_sizes, int n_in,
                              void* d_out, int out_size, void* d_ws, size_t ws_size,
                              hipStream_t stream) {
  (void)in_sizes; (void)n_in; (void)out_size; (void)d_ws; (void)ws_size;
  const float*         x        = (const float*)d_in[0];
  const float*         wfa      = (const float*)d_in[1];
  const int*           rand_idx = (const int*)d_in[2];
  const unsigned char* drop     = (const unsigned char*)d_in[3];  // jnp bool = 1 byte
  float*               out      = (float*)d_out;

  social_group_loss_kernel<<<dim3(kN), dim3(kWaves * 32), 0, stream>>>(
      x, wfa, rand_idx, drop, out);
}